// SparseConvNet_44607530336723
// MI455X (gfx1250) — compile-verified
//
#include <hip/hip_runtime.h>

typedef __attribute__((ext_vector_type(2))) float v2f;
typedef __attribute__((ext_vector_type(8))) float v8f;

#define G3 256
#define VOLN (G3 * G3 * G3)
#define LDSTR 268  // 256 data + 2*4 apron + 4 pad (floats); 268%64=12 -> conflict-free

// ---------------- zero the volume ----------------
__global__ void zero_kernel(float4* __restrict__ p, int n4) {
  int i = blockIdx.x * blockDim.x + threadIdx.x;
  int stride = gridDim.x * blockDim.x;
  float4 z = {0.f, 0.f, 0.f, 0.f};
  for (; i < n4; i += stride) p[i] = z;
}

// ---------------- scatter-add sparse voxels ----------------
__global__ void scatter_kernel(const float* __restrict__ feats,
                               const int* __restrict__ coords,
                               float* __restrict__ vol, int n) {
  int i = blockIdx.x * blockDim.x + threadIdx.x;
  if (i >= n) return;
  int d = coords[3 * i + 0];
  int h = coords[3 * i + 1];
  int w = coords[3 * i + 2];
  unsafeAtomicAdd(&vol[(d << 16) + (h << 8) + w], feats[i]);  // global_atomic_add_f32
}

// ---------------- one separable box-sum pass via banded WMMA ----------------
// Processes 16 "lines" x 256 "positions" in-place. Tile staged in LDS with
// zeroed +-4 aprons so zero-padding needs no conditionals.
// D = A x B (16x16):
//   DATA_IS_A=true  (W pass): A = data (M=line,K=pos),  B = band (K,N=outpos)
//   DATA_IS_A=false (H/D):    A = band (M=outpos,K),    B = data (K,N=line)
template <int STRIDE_HI, int LINE_STR, int POS_STR, bool DATA_IS_A>
__global__ __launch_bounds__(32) void boxpass_kernel(float* __restrict__ vol) {
  __shared__ float tile[16 * LDSTR];

  const int t   = threadIdx.x;        // 0..31, one wave per workgroup
  const int q   = t & 15;             // lane&15: M or N index
  const int kg  = (t >> 4) << 1;      // K group: 0 or 2 (f32 A/B VGPR layout)
  const int hi8 = (t >> 4) << 3;      // D layout: lanes>=16 hold M+8

  const int bHi  = blockIdx.x >> 4;
  const int bLo  = blockIdx.x & 15;
  const int base = bHi * STRIDE_HI + bLo * 16 * LINE_STR;

  // ---- stage 16x256 tile into LDS ----
  if (POS_STR == 1) {
    // pass axis is contiguous: float4 copies
    for (int i = t; i < 16 * 64; i += 32) {
      int line = i >> 6;
      int posq = (i & 63) << 2;
      float4 v = *(const float4*)(vol + base + line * LINE_STR + posq);
      *(float4*)(&tile[line * LDSTR + 4 + posq]) = v;
    }
  } else {
    // lines are 16 consecutive w's: each 16-lane group reads 64B contiguous
    for (int i = t; i < 16 * 256; i += 32) {
      int line = i & 15;
      int pos  = i >> 4;
      tile[line * LDSTR + 4 + pos] = vol[base + line + pos * POS_STR];
    }
  }
  if (t < 16) {
#pragma unroll
    for (int e = 0; e < 4; ++e) {
      tile[t * LDSTR + e]       = 0.0f;  // left apron (pos -4..-1)
      tile[t * LDSTR + 260 + e] = 0.0f;  // right apron (pos 256..259)
      tile[t * LDSTR + 264 + e] = 0.0f;  // pad
    }
  }
  __syncthreads();

  // ---- 6 band-operand register constants (identical for A-band / B-band) ----
  v2f band[6];
#pragma unroll
  for (int tt = 0; tt < 6; ++tt) {
    int rel = 4 * tt - 4;               // K-tile base relative to output block
    int d0  = q - rel - kg;             // in-band iff |d0| <= 2
    int d1  = d0 - 1;
    band[tt].x = (d0 >= -2 && d0 <= 2) ? 1.0f : 0.0f;
    band[tt].y = (d1 >= -2 && d1 <= 2) ? 1.0f : 0.0f;
  }

  const float* myrow = &tile[q * LDSTR + 4];

  for (int j = 0; j < 16; ++j) {
    const int c0 = j << 4;
    v8f acc = {0.f, 0.f, 0.f, 0.f, 0.f, 0.f, 0.f, 0.f};
#pragma unroll
    for (int tt = 0; tt < 6; ++tt) {
      const int p = c0 + 4 * tt - 4 + kg;          // even -> 8B-aligned ds_load_b64
      v2f x = *(const v2f*)(myrow + p);
      if (DATA_IS_A)
        acc = __builtin_amdgcn_wmma_f32_16x16x4_f32(
            false, x, false, band[tt], (short)0, acc, false, false);
      else
        acc = __builtin_amdgcn_wmma_f32_16x16x4_f32(
            false, band[tt], false, x, (short)0, acc, false, false);
    }
    // D layout: element (M = v + hi8, N = q) in acc[v]
#pragma unroll
    for (int v = 0; v < 8; ++v) {
      int g;
      if (DATA_IS_A)  // M=line, N=outpos (outpos contiguous in memory)
        g = base + (v + hi8) * LINE_STR + (c0 + q) * POS_STR;
      else            // M=outpos, N=line (line contiguous in memory)
        g = base + q * LINE_STR + (c0 + v + hi8) * POS_STR;
      vol[g] = acc[v];
    }
  }
}

extern "C" void kernel_launch(void* const* d_in, const int* in_sizes, int n_in,
                              void* d_out, int out_size, void* d_ws, size_t ws_size,
                              hipStream_t stream) {
  const float* feats  = (const float*)d_in[0];
  const int*   coords = (const int*)d_in[1];
  float*       vol    = (float*)d_out;
  const int    n      = in_sizes[0];

  zero_kernel<<<4096, 256, 0, stream>>>((float4*)vol, VOLN / 4);
  scatter_kernel<<<(n + 255) / 256, 256, 0, stream>>>(feats, coords, vol, n);

  // 4096 line-groups per pass, one wave each (16x16 out-tiles x 6 WMMA)
  boxpass_kernel<65536, 256, 1,     true ><<<4096, 32, 0, stream>>>(vol);  // W axis
  boxpass_kernel<65536, 1,   256,   false><<<4096, 32, 0, stream>>>(vol);  // H axis
  boxpass_kernel<256,   1,   65536, false><<<4096, 32, 0, stream>>>(vol);  // D axis
}